// UIFeatureEmbedding_86998857548018
// MI455X (gfx1250) — compile-verified
//
#include <hip/hip_runtime.h>

// CDNA5 / gfx1250: wave32, WMMA 16x16x4 fp32 path (full precision, compute is
// free at this arithmetic intensity; kernel is HBM-gather bound).

typedef float v2f __attribute__((ext_vector_type(2)));
typedef float v8f __attribute__((ext_vector_type(8)));

#define BT 32            // batch rows per block
#define LDSS 388         // 384 cols + 4 pad floats -> conflict-free A reads

__global__ __launch_bounds__(256) void ui_embed_wmma(
    const int* __restrict__ f0, const int* __restrict__ f1,
    const int* __restrict__ f2, const int* __restrict__ f3,
    const int* __restrict__ f4, const int* __restrict__ f5,
    const int* __restrict__ f6, const int* __restrict__ f7,
    const float* __restrict__ e0, const float* __restrict__ e1,
    const float* __restrict__ e2, const float* __restrict__ e3,
    const float* __restrict__ e4, const float* __restrict__ e5,
    const float* __restrict__ e6, const float* __restrict__ e7,
    const float* __restrict__ Wu, const float* __restrict__ Wi,
    float* __restrict__ out)
{
    __shared__ float lds[BT * LDSS];   // 32*388*4 = 49,664 B (< 64 KB static)

    const int tid = threadIdx.x;
    const int b0  = blockIdx.x * BT;

    // ---------------- Phase 1: gather 32 rows x 384 f32 into LDS ----------
    {
        const int*   fptr[8] = {f0,f1,f2,f3,f4,f5,f6,f7};
        const float* eptr[8] = {e0,e1,e2,e3,e4,e5,e6,e7};
        const int    dims[8] = {64,64,32,32,64,64,32,32};
        const int    cb[8]   = {0,64,128,160,192,256,320,352};
        #pragma unroll
        for (int f = 0; f < 8; ++f) {
            const int dim  = dims[f];
            const int d4   = dim >> 2;           // float4 per row
            const int nvec = BT * d4;            // 512 or 256
            const int*   fi = fptr[f];
            const float* ef = eptr[f];
            #pragma unroll
            for (int i = tid; i < nvec; i += 256) {
                const int r   = i / d4;
                const int c   = (i - r * d4) << 2;
                const int idx = fi[b0 + r];
                const float4 v = *(const float4*)(ef + (size_t)idx * dim + c);
                *(float4*)(&lds[r * LDSS + cb[f] + c]) = v;
            }
        }
    }

    // --------- Phase 2: per-wave weight (B-matrix) fragments in VGPRs -----
    const int wave  = tid >> 5;        // == head index 0..7
    const int lane  = tid & 31;
    const int khalf = lane >> 4;       // 0: K pair {0,1}; 1: K pair {2,3}
    const int l16   = lane & 15;

    // head cols of W: Wu heads 0..3, Wi heads 4..7, each (64 out, 48 in)
    const float* W = (wave < 4) ? (Wu + wave * (64 * 48))
                                : (Wi + (wave - 4) * (64 * 48));
    // B[k,n] = W[n, k]  (y = x @ W^T). WMMA B layout (mirrors documented A):
    //   vgpr0: lanes0-15 K=k0+0, lanes16-31 K=k0+2; vgpr1: K=k0+1 / K=k0+3.
    v2f wb[12][4];
    #pragma unroll
    for (int n = 0; n < 4; ++n) {
        const float* wn = W + (n * 16 + l16) * 48 + 2 * khalf;
        #pragma unroll
        for (int k = 0; k < 12; ++k)
            wb[k][n] = *(const v2f*)(wn + k * 4);
    }

    __syncthreads();

    // ---------------- Phase 3: WMMA f32 16x16x4, K = 48 -------------------
    const int acol = wave * 48 + 2 * khalf;  // head's K-slice base in LDS
    #pragma unroll
    for (int sub = 0; sub < 2; ++sub) {
        const float* ap = &lds[(sub * 16 + l16) * LDSS + acol];
        v8f acc[4];
        #pragma unroll
        for (int n = 0; n < 4; ++n)
            acc[n] = (v8f){0.f,0.f,0.f,0.f,0.f,0.f,0.f,0.f};

        #pragma unroll
        for (int k = 0; k < 12; ++k) {
            const v2f a = *(const v2f*)(ap + k * 4);   // A frag, reused x4
            #pragma unroll
            for (int n = 0; n < 4; ++n) {
                acc[n] = __builtin_amdgcn_wmma_f32_16x16x4_f32(
                    false, a, false, wb[k][n], (short)0, acc[n], false, false);
            }
        }

        // store: out[b, head, d]; C/D layout: vgpr v -> M = v + 8*khalf
        #pragma unroll
        for (int n = 0; n < 4; ++n) {
            const int col = wave * 64 + n * 16 + l16;
            #pragma unroll
            for (int v = 0; v < 8; ++v) {
                const int b = b0 + sub * 16 + v + 8 * khalf;
                out[(size_t)b * 512 + col] = acc[n][v];
            }
        }
    }
}

extern "C" void kernel_launch(void* const* d_in, const int* in_sizes, int n_in,
                              void* d_out, int out_size, void* d_ws, size_t ws_size,
                              hipStream_t stream) {
    const int B = in_sizes[0];            // 16384
    dim3 grid(B / BT), block(256);
    ui_embed_wmma<<<grid, block, 0, stream>>>(
        (const int*)d_in[0], (const int*)d_in[1],
        (const int*)d_in[2], (const int*)d_in[3],
        (const int*)d_in[4], (const int*)d_in[5],
        (const int*)d_in[6], (const int*)d_in[7],
        (const float*)d_in[8],  (const float*)d_in[9],
        (const float*)d_in[10], (const float*)d_in[11],
        (const float*)d_in[12], (const float*)d_in[13],
        (const float*)d_in[14], (const float*)d_in[15],
        (const float*)d_in[16], (const float*)d_in[17],
        (float*)d_out);
}